// PhysicsInformedSpline_58858231824566
// MI455X (gfx1250) — compile-verified
//
#include <hip/hip_runtime.h>

typedef __attribute__((ext_vector_type(2))) float v2f;
typedef __attribute__((ext_vector_type(8))) float v8f;

// Cubic B-spline basis, N_CTRL=10, clamped uniform knots
// knots = [0,0,0, 0, 1/7..6/7, 1, 1,1,1]  (14 knots)
// Writes 12 values: N[0..9] = basis, N[10]=N[11]=0 (K padding for WMMA).
__device__ __forceinline__ void bspline_basis10(float t, float* N) {
    const float k[14] = {0.f, 0.f, 0.f, 0.f,
                         1.f/7.f, 2.f/7.f, 3.f/7.f, 4.f/7.f, 5.f/7.f, 6.f/7.f,
                         1.f, 1.f, 1.f, 1.f};
    #pragma unroll
    for (int i = 0; i < 10; ++i)
        N[i] = ((t >= k[i]) && (t < k[i + 1])) ? 1.f : 0.f;
    N[10] = 0.f;
    N[11] = 0.f;
    // Cox-de Boor; in-place ascending update reads only old N[i], N[i+1].
    #pragma unroll
    for (int r = 1; r <= 3; ++r) {
        #pragma unroll
        for (int i = 0; i < 10; ++i) {
            if (i < 10 - r) {
                const float dl = k[i + r] - k[i];          // compile-time constants
                const float dr = k[i + r + 1] - k[i + 1];
                const float il = (dl != 0.f) ? 1.f / dl : 0.f;
                const float ir = (dr != 0.f) ? 1.f / dr : 0.f;
                const float left  = (t - k[i]) * il;
                const float right = (k[i + r + 1] - t) * ir;
                N[i] = left * N[i] + right * N[i + 1];
            } else {
                N[i] = 0.f;
            }
        }
    }
}

// One wave32 per 32 points = two 16-point M-tiles; each lane evaluates the
// basis for exactly one point, halves are exchanged with shfl_xor(16).
// K = 10 -> 3 chunks of 4 (padded to 12). N = 66 -> 5 column tiles of 16.
// fp32 WMMA 16x16x4 (fp32 mandatory: FD derivative with h=1e-6).
__global__ __launch_bounds__(256, 1) void spline_wmma_kernel(
    const float* __restrict__ t, const float* __restrict__ cp,
    float* __restrict__ out, int P) {
    const int lane = threadIdx.x & 31;
    const int wave = threadIdx.x >> 5;
    const int lo = lane & 15;            // column / M-row selector
    const int hi = lane >> 4;            // 0: K pair {0,1}, 1: K pair {2,3}
    const bool h = (hi != 0);
    const long long base = ((long long)blockIdx.x * 8 + wave) * 32;
    const long long Pl = (long long)P;

    // ---- B fragments: B[k][n] = cp[n*10 + k], n = j*3 + kk in [0,66) ----
    // Slots (kb, kb+1) are contiguous and 8B-aligned -> one float2 load each.
    v2f Bf[3][5];
    #pragma unroll
    for (int ct = 0; ct < 5; ++ct) {
        const int n = ct * 16 + lo;
        #pragma unroll
        for (int c = 0; c < 3; ++c) {
            const int kb = 4 * c + 2 * hi;          // 0,2 / 4,6 / 8,10
            v2f b; b.x = 0.f; b.y = 0.f;
            if (n < 66 && kb < 10)                   // kb==10 (c=2,hi=1) is zero pad
                b = *(const v2f*)(cp + n * 10 + kb);
            Bf[c][ct] = b;
        }
    }

    // ---- per-lane basis + FD derivative, one distinct point per lane ----
    long long pidx = base + lane;
    if (pidx > Pl - 1) pidx = Pl - 1;     // tail clamp; keeps loads in range
    const float tv = t[pidx];

    float N0[12], Np[12], Nm[12], D1[12];
    bspline_basis10(tv, N0);
    bspline_basis10(tv + 1e-6f, Np);
    bspline_basis10(tv - 1e-6f, Nm);
    #pragma unroll
    for (int i = 0; i < 12; ++i) D1[i] = (Np[i] - Nm[i]) * 5.0e5f; // /(2h)

    // ---- A fragments for both tiles via lane-xor-16 exchange ----
    // tile0 rows = points base+0..15 (owned by lanes 0..15)
    // tile1 rows = points base+16..31 (owned by lanes 16..31)
    // Lane needs K slots {4c,4c+1} (hi=0) or {4c+2,4c+3} (hi=1) of the row owner.
    v2f A0[2][3], A1[2][3];
    #pragma unroll
    for (int c = 0; c < 3; ++c) {
        const float x0 = N0[4 * c],     x1 = N0[4 * c + 1];
        const float x2 = N0[4 * c + 2], x3 = N0[4 * c + 3];
        const float y0 = D1[4 * c],     y1 = D1[4 * c + 1];
        const float y2 = D1[4 * c + 2], y3 = D1[4 * c + 3];
        const float sx0 = __shfl_xor(x0, 16, 32);
        const float sx1 = __shfl_xor(x1, 16, 32);
        const float sx2 = __shfl_xor(x2, 16, 32);
        const float sx3 = __shfl_xor(x3, 16, 32);
        const float sy0 = __shfl_xor(y0, 16, 32);
        const float sy1 = __shfl_xor(y1, 16, 32);
        const float sy2 = __shfl_xor(y2, 16, 32);
        const float sy3 = __shfl_xor(y3, 16, 32);
        // tile0: lane<16 uses own {x0,x1}; lane>=16 uses partner's {x2,x3}
        A0[0][c].x = h ? sx2 : x0;   A0[0][c].y = h ? sx3 : x1;
        A1[0][c].x = h ? sy2 : y0;   A1[0][c].y = h ? sy3 : y1;
        // tile1: lane<16 uses partner's {x0,x1}; lane>=16 uses own {x2,x3}
        A0[1][c].x = h ? x2 : sx0;   A0[1][c].y = h ? x3 : sx1;
        A1[1][c].x = h ? y2 : sy0;   A1[1][c].y = h ? y3 : sy1;
    }

    const long long P66 = Pl * 66;

    #pragma unroll
    for (int tt = 0; tt < 2; ++tt) {
        // ---- accumulate: 5 column tiles x 3 K-chunks x 2 matrices ----
        v8f Cp[5] = {};
        v8f Cv[5] = {};
        #pragma unroll
        for (int ct = 0; ct < 5; ++ct) {
            #pragma unroll
            for (int c = 0; c < 3; ++c) {
                Cp[ct] = __builtin_amdgcn_wmma_f32_16x16x4_f32(
                    false, A0[tt][c], false, Bf[c][ct], (short)0, Cp[ct], false, false);
                Cv[ct] = __builtin_amdgcn_wmma_f32_16x16x4_f32(
                    false, A1[tt][c], false, Bf[c][ct], (short)0, Cv[ct], false, false);
            }
        }

        // ---- stores: D VGPR r -> row (tbase + r + 8*hi), col (ct*16 + lo) ----
        const long long tbase = base + 16 * tt;
        const long long rowbase = (tbase + 8LL * hi) * 66 + lo;
        float* __restrict__ pp = out + rowbase;                 // pos
        float* __restrict__ vp = out + P66 + rowbase;           // vel
        float* __restrict__ ap = out + 2 * P66 + rowbase;       // acc (== vel)

        if (tbase + 16 <= Pl) {
            // Full tile: tiles 0..3 unpredicated, tile 4 only cols 64,65 valid.
            #pragma unroll
            for (int ct = 0; ct < 4; ++ct) {
                #pragma unroll
                for (int r = 0; r < 8; ++r) {
                    const int off = ct * 16 + r * 66;
                    const float pv = Cp[ct][r];
                    const float vv = Cv[ct][r];
                    pp[off] = pv;
                    vp[off] = vv;
                    ap[off] = vv;
                }
            }
            if (lo < 2) {
                #pragma unroll
                for (int r = 0; r < 8; ++r) {
                    const int off = 64 + r * 66;
                    const float pv = Cp[4][r];
                    const float vv = Cv[4][r];
                    pp[off] = pv;
                    vp[off] = vv;
                    ap[off] = vv;
                }
            }
        } else {
            // Tail tile (unused when P % 32 == 0, kept for safety).
            #pragma unroll
            for (int ct = 0; ct < 5; ++ct) {
                const int n = ct * 16 + lo;
                if (n < 66) {
                    #pragma unroll
                    for (int r = 0; r < 8; ++r) {
                        if (tbase + r + 8 * hi < Pl) {
                            const int off = ct * 16 + r * 66;
                            const float pv = Cp[ct][r];
                            const float vv = Cv[ct][r];
                            pp[off] = pv;
                            vp[off] = vv;
                            ap[off] = vv;
                        }
                    }
                }
            }
        }
    }
}

extern "C" void kernel_launch(void* const* d_in, const int* in_sizes, int n_in,
                              void* d_out, int out_size, void* d_ws, size_t ws_size,
                              hipStream_t stream) {
    const float* t  = (const float*)d_in[0];   // [32*8192] flat
    const float* cp = (const float*)d_in[1];   // [22,3,10] flat = [66][10]
    float* out = (float*)d_out;                // pos | vel | acc, each P*66
    const int P = in_sizes[0];

    const int pointsPerBlock = 8 * 32;         // 8 waves x 2 tiles x 16 points
    const int grid = (P + pointsPerBlock - 1) / pointsPerBlock;
    spline_wmma_kernel<<<grid, 256, 0, stream>>>(t, cp, out, P);
}